// QKNormRoPEKVCacheTestModel_2903397892593
// MI455X (gfx1250) — compile-verified
//
#include <hip/hip_runtime.h>
#include <hip/hip_bf16.h>
#include <stdint.h>

// ---------------------------------------------------------------------------
// Problem constants
// ---------------------------------------------------------------------------
#define NUM_HEADS     32
#define NUM_KV_HEADS  8
#define HEAD_SIZE     128
#define NUM_TOKENS    8192
#define Q_SIZE        (NUM_HEADS * HEAD_SIZE)        // 4096
#define KV_SIZE       (NUM_KV_HEADS * HEAD_SIZE)     // 1024
#define ROW_SIZE      (Q_SIZE + 2 * KV_SIZE)         // 6144
#define NUM_SLOTS     (1024 * 16)                    // 16384
#define CACHE_HALF    ((size_t)NUM_SLOTS * KV_SIZE)  // 16,777,216 floats
#define CACHE_ELEMS   (2 * CACHE_HALF)               // 33,554,432 floats

// d_out layout (floats): q | k | v | kv_cache
#define Q_OFF  ((size_t)0)
#define K_OFF  ((size_t)NUM_TOKENS * Q_SIZE)                 // 33,554,432
#define V_OFF  (K_OFF + (size_t)NUM_TOKENS * KV_SIZE)        // 41,943,040
#define C_OFF  (V_OFF + (size_t)NUM_TOKENS * KV_SIZE)        // 50,331,648

typedef float v4f __attribute__((ext_vector_type(4)));

// ---------------------------------------------------------------------------
// Kernel 1: stream the input kv_cache into the output cache region.
// Grid exactly tiles the data (8192 blocks x 256 threads x 4 v4f), so no
// bounds checks: 4 independent B128 NT loads in flight, then 4 NT stores.
// ---------------------------------------------------------------------------
__global__ __launch_bounds__(256) void cache_copy_kernel(
    const v4f* __restrict__ src, v4f* __restrict__ dst)
{
    const int base = blockIdx.x * 1024 + threadIdx.x;
    v4f val[4];
#pragma unroll
    for (int c = 0; c < 4; ++c)
        val[c] = __builtin_nontemporal_load(&src[base + c * 256]);
#pragma unroll
    for (int c = 0; c < 4; ++c)
        __builtin_nontemporal_store(val[c], &dst[base + c * 256]);
}

// ---------------------------------------------------------------------------
// Kernel 2: one 256-thread block (8 wave32) per token.
//   - async-stage the 24KB qkv row into LDS (6 x global_load_async_to_lds_b128)
//   - threads 0..63 build the per-token cos/sin table (shared by 40 heads)
//   - each wave handles 6 head-rows of 128 floats (4 per lane):
//       hr  0..31 : Q heads (RMS + RoPE)
//       hr 32..39 : K heads (RMS + RoPE)  -> k out + cache[0]
//       hr 40..47 : V heads (copy)        -> v out + cache[1]
//   NeoX pairing d <-> d+64 == shfl_xor(lane, 16) under wave32.
// ---------------------------------------------------------------------------
__global__ __launch_bounds__(256) void qk_rope_kvcache_kernel(
    const float* __restrict__ qkv,
    const int*   __restrict__ positions,
    const int*   __restrict__ slot_mapping,
    const float* __restrict__ q_weight,
    const float* __restrict__ k_weight,
    float*       __restrict__ out)
{
    __shared__ __align__(16) float s_row[ROW_SIZE];   // 24 KB
    __shared__ __align__(16) float s_cos[64];
    __shared__ __align__(16) float s_sin[64];

    const int t    = blockIdx.x;
    const int tid  = threadIdx.x;
    const int lane = tid & 31;
    const int wave = tid >> 5;

    // ---- async stage: 6 chunks of 4KB (256 lanes x 16B) -------------------
    const float* row = qkv + (size_t)t * ROW_SIZE;
#pragma unroll
    for (int c = 0; c < 6; ++c) {
        const int elt = c * 1024 + tid * 4;                 // float index
        uint32_t lds_addr = (uint32_t)(uintptr_t)(&s_row[elt]);
        uint64_t gaddr    = (uint64_t)(uintptr_t)(row + elt);
        asm volatile("global_load_async_to_lds_b128 %0, %1, off"
                     :: "v"(lds_addr), "v"(gaddr) : "memory");
    }

    // ---- per-token RoPE table (overlaps with async copy) ------------------
    const float pos = (float)positions[t];
    if (tid < 64) {
        // inv_freq = 10000^(-j/64) = exp(-j * ln(10000)/64)
        const float f = pos * __expf(-(float)tid * (9.210340371976184f / 64.0f));
        float s, c;
        __sincosf(f, &s, &c);
        s_cos[tid] = c;
        s_sin[tid] = s;
    }

    asm volatile("s_wait_asynccnt 0" ::: "memory");
    __syncthreads();

    const int   slot = slot_mapping[t];
    const int   j4   = (lane & 15) * 4;                   // freq index base
    const float sgn  = (lane < 16) ? -1.0f : 1.0f;

#pragma unroll
    for (int i = 0; i < 6; ++i) {
        const int hr = wave * 6 + i;                      // 0..47, wave-uniform
        v4f x = *(const v4f*)&s_row[hr * HEAD_SIZE + lane * 4];
        v4f res;

        if (hr < 40) {
            // ---- RMS norm over 128 elems (wave32 reduction) ----
            float ss = x.x * x.x + x.y * x.y + x.z * x.z + x.w * x.w;
#pragma unroll
            for (int m = 16; m > 0; m >>= 1)
                ss += __shfl_xor(ss, m, 32);
            const float rn = __frsqrt_rn(ss * (1.0f / 128.0f) + 1e-6f);

            const float* wp = (hr < NUM_HEADS) ? q_weight : k_weight;
            v4f wv = *(const v4f*)&wp[lane * 4];
            v4f xn = x * (rn * wv);

            // ---- NeoX rotate: partner half lives 16 lanes away ----
            v4f po;
            po.x = __shfl_xor(xn.x, 16, 32);
            po.y = __shfl_xor(xn.y, 16, 32);
            po.z = __shfl_xor(xn.z, 16, 32);
            po.w = __shfl_xor(xn.w, 16, 32);

            v4f cv = *(const v4f*)&s_cos[j4];
            v4f sv = *(const v4f*)&s_sin[j4];
            res = xn * cv + (sgn * sv) * po;
        } else {
            res = x;                                      // V passthrough
        }

        // ---- streamed (non-temporal) B128 stores ----
        if (hr < NUM_HEADS) {
            float* p = out + Q_OFF + (size_t)t * Q_SIZE + hr * HEAD_SIZE + lane * 4;
            __builtin_nontemporal_store(res, (v4f*)p);
        } else if (hr < NUM_HEADS + NUM_KV_HEADS) {
            const int hk = hr - NUM_HEADS;
            float* p1 = out + K_OFF + (size_t)t * KV_SIZE + hk * HEAD_SIZE + lane * 4;
            __builtin_nontemporal_store(res, (v4f*)p1);
            float* p2 = out + C_OFF + (size_t)slot * KV_SIZE + hk * HEAD_SIZE + lane * 4;
            __builtin_nontemporal_store(res, (v4f*)p2);
        } else {
            const int hv = hr - NUM_HEADS - NUM_KV_HEADS;
            float* p1 = out + V_OFF + (size_t)t * KV_SIZE + hv * HEAD_SIZE + lane * 4;
            __builtin_nontemporal_store(res, (v4f*)p1);
            float* p2 = out + C_OFF + CACHE_HALF + (size_t)slot * KV_SIZE + hv * HEAD_SIZE + lane * 4;
            __builtin_nontemporal_store(res, (v4f*)p2);
        }
    }
}

// ---------------------------------------------------------------------------
// Launch
// ---------------------------------------------------------------------------
extern "C" void kernel_launch(void* const* d_in, const int* in_sizes, int n_in,
                              void* d_out, int out_size, void* d_ws, size_t ws_size,
                              hipStream_t stream)
{
    const float* qkv          = (const float*)d_in[0];
    const int*   positions    = (const int*)d_in[1];
    const int*   slot_mapping = (const int*)d_in[2];
    const float* q_weight     = (const float*)d_in[3];
    const float* k_weight     = (const float*)d_in[4];
    const float* kv_cache     = (const float*)d_in[5];
    float*       out          = (float*)d_out;

    // 1) copy kv_cache input -> output cache region (33.5M floats = 8.4M v4f,
    //    4 v4f per thread -> 8192 blocks). Must precede the scatter (stream order).
    cache_copy_kernel<<<(int)(CACHE_ELEMS / 4 / 1024), 256, 0, stream>>>(
        (const v4f*)kv_cache, (v4f*)(out + C_OFF));

    // 2) main fused kernel: one block per token, scatters into the cache
    qk_rope_kvcache_kernel<<<NUM_TOKENS, 256, 0, stream>>>(
        qkv, positions, slot_mapping, q_weight, k_weight, out);
}